// PRFR_91001767068092
// MI455X (gfx1250) — compile-verified
//
#include <hip/hip_runtime.h>

#define B_DIM 64
#define T_DIM 512
#define U_DIM 1024
#define TB    128                     // 4 waves/block
#define DT    0.008f
#define RING  64                      // LDS slots per wave
#define AHEAD 48                      // async pipeline depth (<= 63)

typedef __attribute__((ext_vector_type(2))) float v2f;

// ---- fast math helpers (single HW instruction each) ----
__device__ __forceinline__ float fpow(float x, float y) {
  return __builtin_amdgcn_exp2f(y * __builtin_amdgcn_logf(x));
}
__device__ __forceinline__ float frcp(float x) {
  return __builtin_amdgcn_rcpf(x);
}

// ---- CDNA5 async global -> LDS copy (per-lane b64), non-temporal ----
__device__ __forceinline__ void async_ld_b64(unsigned lds_byte, const float* g) {
  asm volatile("global_load_async_to_lds_b64 %0, %1, off th:TH_LOAD_NT"
               :: "v"(lds_byte), "v"((unsigned long long)(size_t)g)
               : "memory");
}
__device__ __forceinline__ void wait_async_slack(void) {
  asm volatile("s_wait_asynccnt %0" :: "n"(AHEAD - 1) : "memory");
}
__device__ __forceinline__ void wait_async_zero(void) {
  asm volatile("s_wait_asynccnt 0" ::: "memory");
}

__global__ __launch_bounds__(TB) void prfr_scan_kernel(
    const float* __restrict__ X,
    const float* __restrict__ P_sigma, const float* __restrict__ P_phi,
    const float* __restrict__ P_eta,   const float* __restrict__ P_beta,
    const float* __restrict__ P_c2c,   const float* __restrict__ P_chill,
    const float* __restrict__ P_cdark, const float* __restrict__ P_bslow,
    const float* __restrict__ P_hcoef, const float* __restrict__ P_haff,
    const float* __restrict__ P_gamma, const float* __restrict__ P_gdark,
    float* __restrict__ out)
{
  // per wave: RING slots * 32 lanes * 8B = 16 KB; 4 waves -> 64 KB static LDS
  __shared__ float xbuf[(TB / 32) * RING * 64];

  const int tid = blockIdx.x * TB + threadIdx.x;   // 0 .. B*U/2-1
  const int pi  = tid & (U_DIM / 2 - 1);           // u-pair index
  const int b   = tid >> 9;
  const int u   = pi * 2;

  // ---- per-unit parameters, two adjacent units per thread (b64 loads) ----
  v2f sigma = *(const v2f*)&P_sigma[u];
  v2f phi   = *(const v2f*)&P_phi[u];
  v2f eta   = *(const v2f*)&P_eta[u];
  v2f beta  = *(const v2f*)&P_beta[u];
  v2f c2c   = *(const v2f*)&P_c2c[u];
  v2f chill = *(const v2f*)&P_chill[u];
  v2f cdark = *(const v2f*)&P_cdark[u];
  v2f bslow = *(const v2f*)&P_bslow[u];
  v2f hcoef = *(const v2f*)&P_hcoef[u];
  v2f haff  = *(const v2f*)&P_haff[u];
  v2f gammv = *(const v2f*)&P_gamma[u];
  v2f gdark = *(const v2f*)&P_gdark[u];

  const float* xg = X   + ((size_t)b * T_DIM) * U_DIM + u;
  float*       og = out + ((size_t)b * T_DIM) * U_DIM + u;

  // ---- per-wave LDS ring base (low 32 bits of generic addr = LDS offset) ----
  const int wave = threadIdx.x >> 5;
  const int lane = threadIdx.x & 31;
  float* slotp = &xbuf[(wave * RING) * 64 + lane * 2];
  const unsigned lds0 = (unsigned)(size_t)slotp;

  // ---- prologue: issue AHEAD async b64 loads for x indices 0..AHEAD-1 ----
  const float* ga = xg;
  #pragma unroll 8
  for (int k = 0; k < AHEAD; ++k) {
    async_ld_b64(lds0 + (unsigned)k * 256u, ga);
    ga += U_DIM;
  }

  // ---- per-element derived constants and initial state ----
  float r[2], p[2], c[2], cs[2], s[2], g[2], gh[2];
  float gam[2], ar[2], ap[2], dteta[2], ac[2], kdt[2], icd[2];
  float acs[2], bcs[2], iha[2], hco[2], chl[2], smax[2], outk[2];

  const v2f x0 = *(const v2f*)xg;

  #pragma unroll
  for (int e = 0; e < 2; ++e) {
    const float sg = sigma[e], ph = phi[e],  et = eta[e],   bt = beta[e];
    const float cc = c2c[e],   ch = chill[e], cd = cdark[e], bs = bslow[e];
    const float hc = hcoef[e], ha = haff[e],  gm = gammv[e], gd = gdark[e];

    const float darkCurrent = fpow(gd, ch) * cc * 0.5f;
    const float gdark_e     = fpow(2.0f * darkCurrent / cc, 1.0f / ch);
    const float cur2ca      = bt * cd / darkCurrent;

    gam[e]   = gm;
    ar[e]    = 1.0f - DT * sg;
    ap[e]    = 1.0f - DT * ph;
    dteta[e] = DT * et;
    ac[e]    = 1.0f - DT * bt;
    kdt[e]   = DT * cur2ca * cc;
    icd[e]   = frcp(cd);
    acs[e]   = 1.0f - DT * bs;
    bcs[e]   = DT * bs;
    iha[e]   = frcp(ha);
    hco[e]   = hc;
    chl[e]   = ch;
    smax[e]  = et / ph * gdark_e * (1.0f + fpow(cd / ha, hc));
    outk[e]  = -0.5f * cc;

    g[e]  = gdark_e;
    s[e]  = gdark_e * et / ph;
    c[e]  = cd;
    cs[e] = cd;
    r[e]  = x0[e] * gm / sg;
    p[e]  = (et + r[e]) / ph;
    gh[e] = fpow(gdark_e, ch);
  }

  // t = 0 output row is zero (g[:,0,:] == 0 in reference)
  __builtin_nontemporal_store((v2f){0.0f, 0.0f}, (v2f*)og);
  og += U_DIM;

  // ---- one scan step for both elements ----
  #define PRFR_STEP(XPV)                                                     \
    do {                                                                     \
      v2f ov;                                                                \
      _Pragma("unroll")                                                      \
      for (int e = 0; e < 2; ++e) {                                          \
        const float xp   = (XPV)[e];                                         \
        const float r_n  = fmaf(gam[e], xp, ar[e] * r[e]);                   \
        const float p_n  = fmaf(ap[e], p[e], fmaf(DT, r[e], dteta[e]));      \
        const float c_n  = fmaf(ac[e], c[e],                                 \
                     kdt[e] * gh[e] * frcp(fmaf(cs[e], icd[e], 1.0f)));      \
        const float cs_n = fmaf(acs[e], cs[e], bcs[e] * c[e]);               \
        const float s_n  = smax[e] *                                         \
                     frcp(1.0f + fpow(c_n * iha[e], hco[e]));                \
        const float g_n  = fmaf(DT, fmaf(-p[e], g[e], s[e]), g[e]);          \
        const float gh_n = fpow(g_n, chl[e]);                                \
        ov[e] = outk[e] * gh_n;                                              \
        r[e] = r_n; p[e] = p_n; c[e] = c_n; cs[e] = cs_n;                    \
        s[e] = s_n; g[e] = g_n; gh[e] = gh_n;                                \
      }                                                                      \
      __builtin_nontemporal_store(ov, (v2f*)og);                             \
      og += U_DIM;                                                           \
    } while (0)

  // ---- main loop: steady-state software pipeline (issue + slack wait) ----
  for (int t = 1; t < T_DIM - AHEAD; ++t) {
    wait_async_slack();                                  // load #t has landed
    const v2f xpv = *(const v2f*)(slotp + ((t - 1) & (RING - 1)) * 64);
    async_ld_b64(lds0 + (unsigned)(((t + AHEAD - 1) & (RING - 1)) * 256), ga);
    ga += U_DIM;                                         // next x row
    PRFR_STEP(xpv);
  }

  // ---- tail: all 511 loads issued; drain once, then no waits/issues ----
  wait_async_zero();
  for (int t = T_DIM - AHEAD; t < T_DIM; ++t) {
    const v2f xpv = *(const v2f*)(slotp + ((t - 1) & (RING - 1)) * 64);
    PRFR_STEP(xpv);
  }
  #undef PRFR_STEP
}

extern "C" void kernel_launch(void* const* d_in, const int* in_sizes, int n_in,
                              void* d_out, int out_size, void* d_ws, size_t ws_size,
                              hipStream_t stream) {
  (void)in_sizes; (void)n_in; (void)d_ws; (void)ws_size; (void)out_size;
  const float* X      = (const float*)d_in[0];
  const float* sigma  = (const float*)d_in[1];
  const float* phi    = (const float*)d_in[2];
  const float* eta    = (const float*)d_in[3];
  const float* beta   = (const float*)d_in[4];
  const float* c2c    = (const float*)d_in[5];
  const float* chill  = (const float*)d_in[6];
  const float* cdark  = (const float*)d_in[7];
  const float* bslow  = (const float*)d_in[8];
  const float* hcoef  = (const float*)d_in[9];
  const float* haff   = (const float*)d_in[10];
  const float* gamma  = (const float*)d_in[11];
  const float* gdark  = (const float*)d_in[12];
  float* out = (float*)d_out;

  dim3 grid((B_DIM * (U_DIM / 2)) / TB);   // 256 blocks
  dim3 block(TB);
  prfr_scan_kernel<<<grid, block, 0, stream>>>(
      X, sigma, phi, eta, beta, c2c, chill, cdark, bslow, hcoef, haff,
      gamma, gdark, out);
}